// AttentionLayer_66090956750993
// MI455X (gfx1250) — compile-verified
//
#include <hip/hip_runtime.h>
#include <math.h>

typedef float v2f __attribute__((ext_vector_type(2)));
typedef float v8f __attribute__((ext_vector_type(8)));

#define B_   16
#define TV_  128
#define TH_  64
#define F_   512
#define H_   512
#define D_   256

#define STILE 16            // s values per workgroup in the attention kernel
#define SQS   132           // padded row stride for beta/q in LDS (132%64==4 -> conflict-free WMMA A reads)

// ---------------------------------------------------------------------------
// C[M,N] = A[M,K] * Bm[K,N] (+ bias[N]) in fp32 using V_WMMA_F32_16X16X4_F32.
// One 16x16 C tile per wave32; 8 waves per 256-thread block.
// ---------------------------------------------------------------------------
__global__ __launch_bounds__(256) void gemm_f32_wmma(
    const float* __restrict__ A, const float* __restrict__ Bm,
    const float* __restrict__ bias, float* __restrict__ C,
    int M, int K, int N)
{
    const int wave = threadIdx.x >> 5;
    const int lane = threadIdx.x & 31;
    const int tiles_n = N >> 4;
    const int tile = blockIdx.x * 8 + wave;
    const int tm = tile / tiles_n;
    const int tn = tile - tm * tiles_n;
    if (tm * 16 >= M) return;           // wave-uniform guard (EXEC stays all-1s)

    const int m0 = tm << 4, n0 = tn << 4;
    const int l15   = lane & 15;
    const int row   = m0 + l15;
    const int col   = n0 + l15;
    const int khalf = (lane >> 4) << 1; // 0 for lanes 0-15, 2 for lanes 16-31

    v8f c = {};
    for (int k = 0; k < K; k += 4) {
        const int kk = k + khalf;
        v2f a = *(const v2f*)(A + (size_t)row * K + kk);
        v2f b;
        b.x = Bm[(size_t)kk * N + col];
        b.y = Bm[(size_t)(kk + 1) * N + col];
        c = __builtin_amdgcn_wmma_f32_16x16x4_f32(
                false, a, false, b, (short)0, c, false, false);
    }

    const float bv = bias ? bias[col] : 0.0f;
    const int rbase = m0 + ((lane >> 4) << 3);
    #pragma unroll
    for (int r = 0; r < 8; ++r)
        C[(size_t)(rbase + r) * N + col] = c[r] + bv;
}

// ---------------------------------------------------------------------------
// One workgroup (512 threads = 16 wave32) per (b, s-tile of 16):
//   stage Wv_b [TV,D] (128 KB) + Uh tile [16,D] + w into LDS (CDNA5: 320 KB/WG)
//   q[s,t]  = sum_d tanh(Uh[s,d] + Wv'[t,d]) * w[d]     (bias folded into Wv')
//   beta    = softmax_t(q)            (one wave per s-row, wave-local)
//   u[s,:]  = beta[s,:] @ v_b[:,:]    (16x128 . 128x512 via fp32 WMMA)
// ---------------------------------------------------------------------------
__global__ __launch_bounds__(512) void attn_score_softmax_ctx(
    const float* __restrict__ Wv,   // [B*TV, D], bias already added
    const float* __restrict__ Uh,   // [B*TH, D]
    const float* __restrict__ wvec, // [D]
    const float* __restrict__ v,    // [B, TV, F]
    float* __restrict__ out)        // [B, TH, F]
{
    __shared__ float sWv[TV_ * D_];     // 128 KB
    __shared__ float sUh[STILE * D_];   //  16 KB
    __shared__ float sw[D_];            //   1 KB
    __shared__ float sq[STILE * SQS];   // ~8.25 KB  (q, then beta)

    const int tid  = threadIdx.x;
    const int wave = tid >> 5;
    const int lane = tid & 31;
    const int bb    = blockIdx.x >> 2;          // TH_/STILE == 4 tiles per batch
    const int s0    = (blockIdx.x & 3) * STILE;

    // ---- stage Wv_b, Uh tile, w into LDS (coalesced float4) ----
    {
        const float4* src = (const float4*)(Wv + (size_t)bb * TV_ * D_);
        float4* dst = (float4*)sWv;
        #pragma unroll
        for (int i = tid; i < (TV_ * D_) / 4; i += 512) dst[i] = src[i];
    }
    {
        const float4* src = (const float4*)(Uh + (size_t)(bb * TH_ + s0) * D_);
        float4* dst = (float4*)sUh;
        #pragma unroll
        for (int i = tid; i < (STILE * D_) / 4; i += 512) dst[i] = src[i];
    }
    if (tid < D_) sw[tid] = wvec[tid];
    __syncthreads();

    // ---- scores: 16*128 = 2048 (s,t) pairs, 128 per wave ----
    for (int p = wave; p < STILE * TV_; p += 16) {
        const int s = p >> 7;            // p / TV_
        const int t = p & (TV_ - 1);
        const float2* wrow = (const float2*)(sWv + t * D_);
        const float2* urow = (const float2*)(sUh + s * D_);
        const float2* wv2  = (const float2*)sw;
        float acc = 0.0f;
        #pragma unroll
        for (int j = 0; j < D_ / 64; ++j) {
            const int d2 = lane + (j << 5);
            const float2 u2 = urow[d2];
            const float2 m2 = wrow[d2];
            const float2 c2 = wv2[d2];
            acc += tanhf(u2.x + m2.x) * c2.x;    // TRANS, co-executes with VALU
            acc += tanhf(u2.y + m2.y) * c2.y;
        }
        #pragma unroll
        for (int off = 16; off > 0; off >>= 1)
            acc += __shfl_xor(acc, off, 32);
        if (lane == 0) sq[s * SQS + t] = acc;
    }
    __syncthreads();

    // ---- softmax: wave w owns row s = w (16 waves, 16 rows) ----
    {
        float* qr = sq + wave * SQS;
        float x[TV_ / 32];
        float mx = -3.4e38f;
        #pragma unroll
        for (int j = 0; j < TV_ / 32; ++j) {
            x[j] = qr[lane + (j << 5)];
            mx = fmaxf(mx, x[j]);
        }
        #pragma unroll
        for (int off = 16; off > 0; off >>= 1)
            mx = fmaxf(mx, __shfl_xor(mx, off, 32));
        float sum = 0.0f;
        #pragma unroll
        for (int j = 0; j < TV_ / 32; ++j) {
            x[j] = expf(x[j] - mx);
            sum += x[j];
        }
        #pragma unroll
        for (int off = 16; off > 0; off >>= 1)
            sum += __shfl_xor(sum, off, 32);
        const float inv = 1.0f / sum;
        #pragma unroll
        for (int j = 0; j < TV_ / 32; ++j)
            qr[lane + (j << 5)] = x[j] * inv;    // beta row, in place
    }
    __syncthreads();

    // ---- context GEMM: [16 x 128] beta (LDS) x [128 x 512] v (L2) via WMMA ----
    const float* vB = v + (size_t)bb * TV_ * F_;
    const int l15   = lane & 15;
    const int khalf = (lane >> 4) << 1;
    const int rbase = (lane >> 4) << 3;
    for (int tile = wave; tile < F_ / 16; tile += 16) {   // 2 tiles per wave
        const int f0 = tile << 4;
        v8f c = {};
        for (int k = 0; k < TV_; k += 4) {
            const int kk = k + khalf;
            // A-frag: beta[l15][kk..kk+1] from LDS (stride 132 -> bank-conflict-free)
            v2f a = *(const v2f*)(sq + l15 * SQS + kk);
            v2f b;
            b.x = vB[(size_t)kk * F_ + f0 + l15];
            b.y = vB[(size_t)(kk + 1) * F_ + f0 + l15];
            c = __builtin_amdgcn_wmma_f32_16x16x4_f32(
                    false, a, false, b, (short)0, c, false, false);
        }
        #pragma unroll
        for (int r = 0; r < 8; ++r)
            out[(size_t)(bb * TH_ + s0 + rbase + r) * F_ + f0 + l15] = c[r];
    }
}

// ---------------------------------------------------------------------------
extern "C" void kernel_launch(void* const* d_in, const int* in_sizes, int n_in,
                              void* d_out, int out_size, void* d_ws, size_t ws_size,
                              hipStream_t stream) {
    (void)in_sizes; (void)n_in; (void)out_size; (void)ws_size;
    const float* v    = (const float*)d_in[0];  // [B, TV, F]
    const float* h    = (const float*)d_in[1];  // [B, TH, H]
    const float* W    = (const float*)d_in[2];  // [F, D]
    const float* U    = (const float*)d_in[3];  // [H, D]
    const float* bvec = (const float*)d_in[4];  // [D]
    const float* wvec = (const float*)d_in[5];  // [D, 1]
    float* out = (float*)d_out;                 // [B, TH, F]

    float* Wv = (float*)d_ws;                   // [B*TV, D] = 2 MB
    float* Uh = Wv + (size_t)B_ * TV_ * D_;     // [B*TH, D] = 1 MB

    // Wv = v @ W + b   : M=2048, K=512, N=256 -> 2048 tiles / 8 waves = 256 blocks
    gemm_f32_wmma<<<256, 256, 0, stream>>>(v, W, bvec, Wv, B_ * TV_, F_, D_);
    // Uh = h @ U       : M=1024 -> 128 blocks
    gemm_f32_wmma<<<128, 256, 0, stream>>>(h, U, nullptr, Uh, B_ * TH_, H_, D_);
    // fused tanh-score + softmax + WMMA context: one block per (b, s-tile)
    attn_score_softmax_ctx<<<B_ * (TH_ / STILE), 512, 0, stream>>>(Wv, Uh, wvec, v, out);
}